// W_SimVQ_decompose_19765439496214
// MI455X (gfx1250) — compile-verified
//
#include <hip/hip_runtime.h>

// ---------------- problem constants (from the reference) ----------------
#define N_ROWS 65536   // 1024*64 flattened tokens
#define N_EMBED 4096   // codebook entries
#define DIM 512        // feature dim
#define BETA 0.25f

// ---------------- tiling ----------------
#define MT 128         // rows per workgroup tile
#define NT 128         // cols per workgroup tile
#define KC 16          // K-chunk staged in LDS
#define KCP 20         // padded LDS row pitch (floats) — kills ds bank conflicts

typedef float v2f __attribute__((ext_vector_type(2)));
typedef float v8f __attribute__((ext_vector_type(8)));

// D = A(16x4) * B(4x16) + C, fp32 WMMA (CDNA5; 8-arg builtin form)
__device__ __forceinline__ v8f wmma_f32(v2f a, v2f b, v8f c) {
  return __builtin_amdgcn_wmma_f32_16x16x4_f32(false, a, false, b, (short)0, c,
                                               false, false);
}

// ---- synchronous staging (used by the small codebook GEMM) ----
// 128 rows x 16 cols fp32: src[row_base+r][kc..kc+15] -> lds[r*KCP + c]
__device__ __forceinline__ void stage_tile(const float* __restrict__ src,
                                           int row_base, int kc,
                                           float* lds, int tid) {
#pragma unroll
  for (int p = 0; p < 2; ++p) {
    const int r = p * 64 + (tid >> 2);
    const int cv = (tid & 3) * 4;
    const float4 v =
        *(const float4*)(src + (size_t)(row_base + r) * DIM + kc + cv);
    *(float4*)(lds + r * KCP + cv) = v;
  }
}

// ---- CDNA5 async staging: GLOBAL_LOAD_ASYNC_TO_LDS_B128 (ASYNCcnt) ----
// Same 128x16 tile; per-lane 16B global->LDS copies, no VGPR round-trip.
__device__ __forceinline__ void stage_tile_async(const float* __restrict__ src,
                                                 int row_base, int kc,
                                                 float* lds, int tid) {
#pragma unroll
  for (int p = 0; p < 2; ++p) {
    const int r = p * 64 + (tid >> 2);
    const int cv = (tid & 3) * 4;
    const unsigned long long g =
        (unsigned long long)(size_t)(src + (size_t)(row_base + r) * DIM + kc + cv);
    const unsigned l = (unsigned)(size_t)(lds + r * KCP + cv);  // LDS byte offset
    asm volatile("global_load_async_to_lds_b128 %0, %1, off"
                 :
                 : "v"(l), "v"(g)
                 : "memory");
  }
}

__device__ __forceinline__ void wait_async0() {
  asm volatile("s_wait_asynccnt 0x0" ::: "memory");
}

// ---------------------------------------------------------------
// Kernel 1: cb[e][j] = sum_d emb[e][d]*W[j][d] + b[j]   (A·B^T + bias)
// grid (N_EMBED/MT, DIM/NT) x 256 threads
// ---------------------------------------------------------------
__global__ __launch_bounds__(256) void k_cb_gemm(const float* __restrict__ emb,
                                                 const float* __restrict__ pw,
                                                 const float* __restrict__ pb,
                                                 float* __restrict__ cb) {
  __shared__ float As[MT * KCP];
  __shared__ float Bs[NT * KCP];
  const int tid = threadIdx.x;
  const int wave = tid >> 5;
  const int lane = tid & 31;
  const int l16 = lane & 15;
  const int lhi = lane >> 4;
  const int mbase = blockIdx.x * MT;
  const int nbase = blockIdx.y * NT;

  const v8f vzero = {0.f, 0.f, 0.f, 0.f, 0.f, 0.f, 0.f, 0.f};
  v8f acc[8];
#pragma unroll
  for (int n = 0; n < 8; ++n) acc[n] = vzero;

  for (int kc = 0; kc < DIM; kc += KC) {
    __syncthreads();
    stage_tile(emb, mbase, kc, As, tid);
    stage_tile(pw, nbase, kc, Bs, tid);
    __syncthreads();
#pragma unroll
    for (int kk = 0; kk < KC; kk += 4) {
      const v2f a = *(const v2f*)(&As[(wave * 16 + l16) * KCP + kk + 2 * lhi]);
#pragma unroll
      for (int n = 0; n < 8; ++n) {
        const v2f b = *(const v2f*)(&Bs[(n * 16 + l16) * KCP + kk + 2 * lhi]);
        acc[n] = wmma_f32(a, b, acc[n]);
      }
    }
  }
#pragma unroll
  for (int n = 0; n < 8; ++n) {
    const int col = nbase + n * 16 + l16;
    const float bias = pb[col];
#pragma unroll
    for (int r = 0; r < 8; ++r) {
      const int row = mbase + wave * 16 + r + 8 * lhi;  // C layout: VGPR r
      cb[(size_t)row * DIM + col] = acc[n][r] + bias;
    }
  }
}

// ---------------------------------------------------------------
// Row squared-norms: one wave per 512-float row. grid = rows/8 x 256
// ---------------------------------------------------------------
__global__ __launch_bounds__(256) void k_rownorm(const float* __restrict__ m,
                                                 float* __restrict__ out) {
  const int row = blockIdx.x * 8 + (threadIdx.x >> 5);
  const int lane = threadIdx.x & 31;
  const float* p = m + (size_t)row * DIM;
  float s = 0.f;
#pragma unroll
  for (int j = 0; j < DIM / 32; ++j) {
    const float v = p[lane + j * 32];
    s += v * v;
  }
#pragma unroll
  for (int msk = 16; msk >= 1; msk >>= 1) s += __shfl_xor(s, msk, 32);
  if (lane == 0) out[row] = s;
}

// ---------------------------------------------------------------
// Kernel 2: d[i][e] = x2[i] + c2[e] - 2*flat[i]·cb[e]; fused per-row argmin.
// Workgroup: 128 rows x all 4096 cols. 8 waves as 4(M)x2(N); each wave
// computes a 32x64 sub-tile (2x4 register blocking -> 6 ds-b64 per 8 WMMA).
// Global->LDS staging uses async-to-LDS with double buffering (ASYNCcnt).
// grid = N_ROWS/MT x 256 threads
// ---------------------------------------------------------------
__global__ __launch_bounds__(256) void k_dist_argmin(
    const float* __restrict__ flat, const float* __restrict__ cb,
    const float* __restrict__ x2, const float* __restrict__ c2,
    float* __restrict__ dmat, int* __restrict__ amin) {
  __shared__ float As[2][MT * KCP];
  __shared__ float Bs[2][NT * KCP];
  __shared__ float sval[2][MT];
  __shared__ int sidx[2][MT];

  const int tid = threadIdx.x;
  const int wave = tid >> 5;
  const int lane = tid & 31;
  const int l16 = lane & 15;
  const int lhi = lane >> 4;
  const int wave_m = wave & 3;   // 0..3 -> 32-row strip
  const int wave_n = wave >> 2;  // 0..1 -> 64-col strip
  const int mbase = blockIdx.x * MT;

  float xr[2][8];
#pragma unroll
  for (int mi = 0; mi < 2; ++mi)
#pragma unroll
    for (int r = 0; r < 8; ++r)
      xr[mi][r] = x2[mbase + wave_m * 32 + mi * 16 + r + 8 * lhi];

  float bestv[2][8];
  int beste[2][8];
#pragma unroll
  for (int mi = 0; mi < 2; ++mi)
#pragma unroll
    for (int r = 0; r < 8; ++r) {
      bestv[mi][r] = 3.402823466e38f;
      beste[mi][r] = 0;
    }

  const v8f vzero = {0.f, 0.f, 0.f, 0.f, 0.f, 0.f, 0.f, 0.f};
  const int nchunks = DIM / KC;  // 32

  for (int ct = 0; ct < N_EMBED; ct += NT) {
    v8f acc[2][4];
#pragma unroll
    for (int mi = 0; mi < 2; ++mi)
#pragma unroll
      for (int ni = 0; ni < 4; ++ni) acc[mi][ni] = vzero;

    // prologue: async-fill buffer 0
    stage_tile_async(flat, mbase, 0, As[0], tid);
    stage_tile_async(cb, ct, 0, Bs[0], tid);
    wait_async0();
    __syncthreads();

    for (int c = 0; c < nchunks; ++c) {
      const int cur = c & 1;
      if (c + 1 < nchunks) {  // overlap next chunk's HBM/L2 fetch with WMMA
        stage_tile_async(flat, mbase, (c + 1) * KC, As[cur ^ 1], tid);
        stage_tile_async(cb, ct, (c + 1) * KC, Bs[cur ^ 1], tid);
      }
      const float* Ac = As[cur];
      const float* Bc = Bs[cur];
#pragma unroll
      for (int kk = 0; kk < KC; kk += 4) {
        v2f a[2], b[4];
#pragma unroll
        for (int mi = 0; mi < 2; ++mi)
          a[mi] = *(const v2f*)(&Ac[(wave_m * 32 + mi * 16 + l16) * KCP + kk +
                                    2 * lhi]);
#pragma unroll
        for (int ni = 0; ni < 4; ++ni)
          b[ni] = *(const v2f*)(&Bc[(wave_n * 64 + ni * 16 + l16) * KCP + kk +
                                    2 * lhi]);
#pragma unroll
        for (int mi = 0; mi < 2; ++mi)
#pragma unroll
          for (int ni = 0; ni < 4; ++ni)
            acc[mi][ni] = wmma_f32(a[mi], b[ni], acc[mi][ni]);
      }
      wait_async0();    // next buffer landed in LDS
      __syncthreads();  // everyone done reading cur; next is visible
    }

    // epilogue: distances + streaming d store + running argmin
#pragma unroll
    for (int ni = 0; ni < 4; ++ni) {
      const int col = ct + wave_n * 64 + ni * 16 + l16;
      const float c2v = c2[col];
#pragma unroll
      for (int mi = 0; mi < 2; ++mi)
#pragma unroll
        for (int r = 0; r < 8; ++r) {
          const int row = mbase + wave_m * 32 + mi * 16 + r + 8 * lhi;
          const float dv = xr[mi][r] + c2v - 2.0f * acc[mi][ni][r];
          dmat[((size_t)row << 12) + col] = dv;  // N_EMBED == 4096 == 1<<12
          if (dv < bestv[mi][r]) {  // strict < keeps earliest col
            bestv[mi][r] = dv;
            beste[mi][r] = col;
          }
        }
    }
  }

  // cross-lane argmin within each 16-lane half (rows differ by half)
#pragma unroll
  for (int mi = 0; mi < 2; ++mi)
#pragma unroll
    for (int r = 0; r < 8; ++r) {
#pragma unroll
      for (int msk = 8; msk >= 1; msk >>= 1) {
        const float ov = __shfl_xor(bestv[mi][r], msk, 32);
        const int oe = __shfl_xor(beste[mi][r], msk, 32);
        if (ov < bestv[mi][r] || (ov == bestv[mi][r] && oe < beste[mi][r])) {
          bestv[mi][r] = ov;
          beste[mi][r] = oe;
        }
      }
      if (l16 == 0) {
        const int rl = wave_m * 32 + mi * 16 + r + 8 * lhi;  // 0..127
        sval[wave_n][rl] = bestv[mi][r];
        sidx[wave_n][rl] = beste[mi][r];
      }
    }
  __syncthreads();
  // combine the two column-halves (wave_n = 0,1), lowest-index tie-break
  if (tid < MT) {
    float v0 = sval[0][tid];
    int i0 = sidx[0][tid];
    const float v1 = sval[1][tid];
    const int i1 = sidx[1][tid];
    if (v1 < v0 || (v1 == v0 && i1 < i0)) {
      v0 = v1;
      i0 = i1;
    }
    amin[mbase + tid] = i0;
  }
}

// ---------------------------------------------------------------
// Kernel 3: gather z_q (== z_quantize numerically), embed_ind (as float),
// per-row sum (z_q - x)^2. grid = N_ROWS x 256
// ---------------------------------------------------------------
__global__ __launch_bounds__(256) void k_gather(
    const float* __restrict__ flat, const float* __restrict__ cb,
    const int* __restrict__ amin, float* __restrict__ zq,
    float* __restrict__ ind_out, float* __restrict__ rowsum) {
  const int row = blockIdx.x;
  const int tid = threadIdx.x;
  const int e = amin[row];
  __shared__ float red[8];
  float s = 0.f;
#pragma unroll
  for (int j = 0; j < DIM / 256; ++j) {
    const int c = tid + j * 256;
    const float z = cb[(size_t)e * DIM + c];
    const float xv = flat[(size_t)row * DIM + c];
    zq[(size_t)row * DIM + c] = z;
    const float dd = z - xv;
    s += dd * dd;
  }
#pragma unroll
  for (int msk = 16; msk >= 1; msk >>= 1) s += __shfl_xor(s, msk, 32);
  if ((tid & 31) == 0) red[tid >> 5] = s;
  __syncthreads();
  if (tid == 0) {
    float t = 0.f;
#pragma unroll
    for (int w = 0; w < 8; ++w) t += red[w];
    rowsum[row] = t;
    ind_out[row] = (float)e;
  }
}

// ---------------------------------------------------------------
// Kernel 4: deterministic final reduction -> diff = (1+BETA)*mean
// ---------------------------------------------------------------
__global__ __launch_bounds__(256) void k_diff(const float* __restrict__ rowsum,
                                              float* __restrict__ diff) {
  __shared__ float red[8];
  const int tid = threadIdx.x;
  float s = 0.f;
  for (int i = tid; i < N_ROWS; i += 256) s += rowsum[i];
#pragma unroll
  for (int msk = 16; msk >= 1; msk >>= 1) s += __shfl_xor(s, msk, 32);
  if ((tid & 31) == 0) red[tid >> 5] = s;
  __syncthreads();
  if (tid == 0) {
    float t = 0.f;
#pragma unroll
    for (int w = 0; w < 8; ++w) t += red[w];
    diff[0] = (1.0f + BETA) * (t / (float)((size_t)N_ROWS * DIM));
  }
}

// ---------------------------------------------------------------
extern "C" void kernel_launch(void* const* d_in, const int* in_sizes, int n_in,
                              void* d_out, int out_size, void* d_ws,
                              size_t ws_size, hipStream_t stream) {
  (void)in_sizes; (void)n_in; (void)out_size; (void)ws_size;

  const float* x = (const float*)d_in[0];    // [65536,512]
  const float* emb = (const float*)d_in[1];  // [4096,512]
  const float* pw = (const float*)d_in[2];   // [512,512]
  const float* pb = (const float*)d_in[3];   // [512]
  // d_in[4] = is_look_back (no-op in reference)

  // workspace layout
  char* ws = (char*)d_ws;
  float* cb = (float*)ws;                                // 4096*512
  float* c2 = (float*)(ws + (size_t)N_EMBED * DIM * 4);  // 4096
  float* x2 = c2 + N_EMBED;                              // 65536
  float* rowsum = x2 + N_ROWS;                           // 65536
  int* amin = (int*)(rowsum + N_ROWS);                   // 65536

  // output layout: z_quantize | diff | embed_ind | d
  float* out = (float*)d_out;
  float* zq = out;
  float* diff = out + (size_t)N_ROWS * DIM;
  float* ind = diff + 1;
  float* dmat = ind + N_ROWS;

  k_cb_gemm<<<dim3(N_EMBED / MT, DIM / NT), 256, 0, stream>>>(emb, pw, pb, cb);
  k_rownorm<<<N_EMBED / 8, 256, 0, stream>>>(cb, c2);
  k_rownorm<<<N_ROWS / 8, 256, 0, stream>>>(x, x2);
  k_dist_argmin<<<N_ROWS / MT, 256, 0, stream>>>(x, cb, x2, c2, dmat, amin);
  k_gather<<<N_ROWS, 256, 0, stream>>>(x, cb, amin, zq, ind, rowsum);
  k_diff<<<1, 256, 0, stream>>>(rowsum, diff);
}